// ByteMulSwiGLU_7945689497933
// MI455X (gfx1250) — compile-verified
//
#include <hip/hip_runtime.h>
#include <math.h>

// ---------------------------------------------------------------------------
// ByteMulSwiGLU for MI455X (gfx1250, wave32).
//
// Only row 0 of the 4x64 GE matrix affects the output, and row 0 has exactly
// 3 non-zeros (byte_a@0, byte_b@1, op@29). Layer-1 hiddens are computed
// sparsely on the VALU; the dense GEMMs run on V_WMMA_F32_16X16X4_F32
// (exact f32 semantics — the result is round(v)&255, so reduced precision
// would flip bytes near .5 boundaries).
//
// v2: weight matrices are stored in LDS pre-swizzled into per-lane fragment
// order, so every WMMA B fragment is one aligned ds_load_b64 directly into
// an even VGPR pair (no v_mov repacking in the inner loop).
// ---------------------------------------------------------------------------

#define BD        256
#define T_TOK     65536      // B*S = 8*8192
#define N_TILES   4096       // T_TOK / 16

typedef __attribute__((ext_vector_type(2))) float v2f;
typedef __attribute__((ext_vector_type(8))) float v8f;

__device__ __forceinline__ v8f wmma4(v2f a, v2f b, v8f c) {
  // (neg_a, A, neg_b, B, c_mod, C, reuse_a, reuse_b)
  return __builtin_amdgcn_wmma_f32_16x16x4_f32(
      false, a, false, b, (short)0, c, false, false);
}

// Wave-private LDS producer->consumer sync.
__device__ __forceinline__ void wave_lds_sync() {
  asm volatile("s_wait_dscnt 0x0" ::: "memory");
  __builtin_amdgcn_wave_barrier();
}

__device__ __forceinline__ float silu_mul(float h1, float h2) {
  return (h1 / (1.0f + expf(-h1))) * h2;
}

// Swizzled index for WMMA B fragments: element (k, n) of a KxN weight matrix
// (NT = N/16 n-tiles). Lane l of fragment (k0=4*(k>>2), nt) reads the v2f at
// float index (((k>>2)*NT + nt)*32 + l)*2, getting K = k0+2*(l>>4) (+1).
__device__ __forceinline__ int bswz(int k, int n, int NT) {
  int g = k >> 2, kk = k & 3, nt = n >> 4, N = n & 15;
  return (((g * NT + nt) * 32) + ((kk >> 1) * 16) + N) * 2 + (kk & 1);
}

// index of first element > 0.5 in 16 consecutive floats, 0 if none
__device__ __forceinline__ float first_hot16(const float* __restrict__ p) {
  int idx = 0;
  bool found = false;
#pragma unroll
  for (int k = 0; k < 16; ++k) {
    float v = p[k];
    if (!found && v > 0.5f) { idx = k; found = true; }
  }
  return (float)idx;
}

// ---------------------------------------------------------------------------
// Kernel 1: decode + MLP, one 16-token tile per wave.
// Dynamic LDS layout (floats):
//   [0      ) sW3s  : W3_0 swizzled        128*64 = 8192
//   [8192   ) sW11s : W1_1 swizzled         64*128 = 8192
//   [16384  ) sW21s : W2_1 swizzled         64*128 = 8192
//   [24576  ) sW12r : rows {0,1,29} of W1_0 then W2_0   6*128 = 768
//   [25344  ) sW31c : W3_1[:,40]            128
//   [25472  ) 8 per-wave arenas of 3392 floats:
//              G 16x128 (2048) | Y 16x64 (1024) | P 16x16 (256) | AB 16x4 (64)
// Total 52608 floats = 210,432 bytes (< 320 KB WGP LDS).
// ---------------------------------------------------------------------------
__global__ __launch_bounds__(256) void
bms_compute(const float* __restrict__ x,
            const float* __restrict__ W1_0, const float* __restrict__ W2_0,
            const float* __restrict__ W3_0, const float* __restrict__ W1_1,
            const float* __restrict__ W2_1, const float* __restrict__ W3_1,
            int* __restrict__ ws) {
  extern __shared__ float sm[];
  float* sW3s  = sm;
  float* sW11s = sm + 8192;
  float* sW21s = sm + 16384;
  float* sW12r = sm + 24576;
  float* sW31c = sm + 25344;
  float* arena = sm + 25472 + (threadIdx.x >> 5) * 3392;
  float* sG = arena;            // 16 x 128
  float* sY = arena + 2048;     // 16 x 64
  float* sP = arena + 3072;     // 16 x 16 partial sums
  float* sAB = arena + 3328;    // 16 x {a, b, op, mask}

  const int tid  = threadIdx.x;
  const int lane = tid & 31;
  const int wid  = blockIdx.x * 8 + (tid >> 5);   // 4096 waves == 4096 tiles
  const int token0 = wid * 16;

  // Prefetch this wave's token rows (bytes [0,448) of each row) so the
  // decode loads overlap with weight staging.  -> global_prefetch_b8
  if (lane < 16) {
    const float* p = x + (size_t)(token0 + lane) * BD;
    __builtin_prefetch(p, 0, 1);
    __builtin_prefetch(p + 32, 0, 1);
    __builtin_prefetch(p + 64, 0, 1);
    __builtin_prefetch(p + 96, 0, 1);
  }

  // ---- cooperative weight staging into LDS, swizzled for B fragments ----
  for (int i = tid; i < 8192; i += 256) {
    {                                    // W3_0: 128x64, 4 n-tiles
      int k = i >> 6, n = i & 63;
      sW3s[bswz(k, n, 4)] = W3_0[i];
    }
    {                                    // W1_1 / W2_1: 64x128, 8 n-tiles
      int k = i >> 7, n = i & 127;
      int s = bswz(k, n, 8);
      sW11s[s] = W1_1[i];
      sW21s[s] = W2_1[i];
    }
  }
  for (int i = tid; i < 768; i += 256) {
    const int rowmap[3] = {0, 1, 29};  // NIB_A, NIB_B, OP_START+OPCODE
    int r = i >> 7, j = i & 127;
    const float* Wsrc = (r < 3) ? W1_0 : W2_0;
    sW12r[i] = Wsrc[rowmap[r % 3] * 128 + j];
  }
  if (tid < 128) sW31c[tid] = W3_1[tid * 64 + 40];  // RESULT column
  __syncthreads();

  // hoisted sparse-layer weights: rows {0,1,29} of W1_0/W2_0, col j=lane+32c
  float w1a[4], w1b[4], w1o[4], w2a[4], w2b[4], w2o[4];
#pragma unroll
  for (int c = 0; c < 4; ++c) {
    int j = lane + 32 * c;
    w1a[c] = sW12r[0 * 128 + j];
    w1b[c] = sW12r[1 * 128 + j];
    w1o[c] = sW12r[2 * 128 + j];
    w2a[c] = sW12r[3 * 128 + j];
    w2b[c] = sW12r[4 * 128 + j];
    w2o[c] = sW12r[5 * 128 + j];
  }

  // ---- decode: lanes 0..15, one token each ----
  if (lane < 16) {
    const float* p = x + (size_t)(token0 + lane) * BD;
    float m0 = p[0], m1 = p[1];                          // MARK_AX, OP_MUL
    float a  = first_hot16(p + 16) + 16.0f * first_hot16(p + 32);  // ALU
    float b  = first_hot16(p + 48) + 16.0f * first_hot16(p + 64);  // AX_CARRY
    float op = p[107];                                   // OPCODE_BASE + 27
    sAB[lane * 4 + 0] = a;
    sAB[lane * 4 + 1] = b;
    sAB[lane * 4 + 2] = op;
    sAB[lane * 4 + 3] = (m0 > 0.5f && m1 > 0.5f) ? 1.0f : 0.0f;
  }
  wave_lds_sync();

  // ---- layer-1 hiddens (3-sparse input) + SwiGLU gate -> G in LDS ----
  for (int t = 0; t < 16; ++t) {
    float a = sAB[t * 4 + 0], b = sAB[t * 4 + 1], op = sAB[t * 4 + 2];
#pragma unroll
    for (int c = 0; c < 4; ++c) {
      int j = lane + 32 * c;
      float h1 = a * w1a[c] + b * w1b[c] + op * w1o[c];
      float h2 = a * w2a[c] + b * w2b[c] + op * w2o[c];
      sG[t * 128 + j] = silu_mul(h1, h2);
    }
  }
  wave_lds_sync();

  const int Ml   = lane & 15;         // A-row / C-col index
  const int koff = (lane >> 4) * 2;   // K sub-offset for upper half-wave
  const int mhi  = (lane >> 4) * 8;   // C-row offset for upper half-wave

  // ---- GEMM1: Y(16x64) = G(16x128) @ W3_0(128x64) ----
  v8f acc[4];
#pragma unroll
  for (int n = 0; n < 4; ++n)
    acc[n] = (v8f){0.f, 0.f, 0.f, 0.f, 0.f, 0.f, 0.f, 0.f};
  for (int k0 = 0; k0 < 128; k0 += 4) {
    v2f a = *(const v2f*)&sG[Ml * 128 + k0 + koff];          // ds_load_b64
#pragma unroll
    for (int n = 0; n < 4; ++n) {
      v2f b = *(const v2f*)&sW3s[(((k0 >> 2) * 4 + n) * 32 + lane) * 2];
      acc[n] = wmma4(a, b, acc[n]);
    }
  }
#pragma unroll
  for (int n = 0; n < 4; ++n)
#pragma unroll
    for (int r = 0; r < 8; ++r)
      sY[(r + mhi) * 64 + n * 16 + Ml] = acc[n][r];
  wave_lds_sync();

  // ---- GEMM2: H' = Y @ {W1_1,W2_1}, fused SwiGLU + dot with W3_1[:,40] ----
  float vpart[8];
#pragma unroll
  for (int r = 0; r < 8; ++r) vpart[r] = 0.0f;
  for (int nt = 0; nt < 8; ++nt) {
    v8f a1 = (v8f){0.f, 0.f, 0.f, 0.f, 0.f, 0.f, 0.f, 0.f};
    v8f a2 = (v8f){0.f, 0.f, 0.f, 0.f, 0.f, 0.f, 0.f, 0.f};
    float wc = sW31c[nt * 16 + Ml];
    for (int k0 = 0; k0 < 64; k0 += 4) {
      v2f a  = *(const v2f*)&sY[Ml * 64 + k0 + koff];        // ds_load_b64
      int s  = (((k0 >> 2) * 8 + nt) * 32 + lane) * 2;
      v2f b1 = *(const v2f*)&sW11s[s];
      v2f b2 = *(const v2f*)&sW21s[s];
      a1 = wmma4(a, b1, a1);
      a2 = wmma4(a, b2, a2);
    }
#pragma unroll
    for (int r = 0; r < 8; ++r)
      vpart[r] += silu_mul(a1[r], a2[r]) * wc;
  }
#pragma unroll
  for (int r = 0; r < 8; ++r)
    sP[(r + mhi) * 16 + Ml] = vpart[r];
  wave_lds_sync();

  // ---- fixed-order (deterministic) reduction + RNE byte, pack with mask ----
  if (lane < 16) {
    float v = 0.0f;
#pragma unroll
    for (int n = 0; n < 16; ++n) v += sP[lane * 16 + n];
    int rb = ((int)rintf(v)) & 255;   // matches jnp.round (RNE) + int32 & 255
    int m  = sAB[lane * 4 + 3] > 0.5f ? 1 : 0;
    ws[token0 + lane] = rb | (m << 8);
  }
}

// ---------------------------------------------------------------------------
// Kernel 2: out = x_bd + masked one-hot bumps. Pure bandwidth (float4).
// float4 index i: token = i>>6, quad-in-row = i&63; cols [128,160) are
// quads [32,40).
// ---------------------------------------------------------------------------
__global__ __launch_bounds__(256) void
bms_apply(const float* __restrict__ x, const int* __restrict__ ws,
          float* __restrict__ out) {
  const long N4 = (long)T_TOK * 64;
  long i = (long)blockIdx.x * 256 + threadIdx.x;
  const long stride = (long)gridDim.x * 256;
  const float4* x4 = (const float4*)x;
  float4* o4 = (float4*)out;
  for (; i < N4; i += stride) {
    float4 v = x4[i];
    int c4 = (int)(i & 63);
    if (c4 >= 32 && c4 < 40) {
      int r = ws[(int)(i >> 6)];
      if (r & 256) {
        int tgtLo = 128 + (r & 15);
        int tgtHi = 144 + ((r >> 4) & 15);
        int col0 = c4 * 4;
        if (col0 + 0 == tgtLo || col0 + 0 == tgtHi) v.x += 2.0f;
        if (col0 + 1 == tgtLo || col0 + 1 == tgtHi) v.y += 2.0f;
        if (col0 + 2 == tgtLo || col0 + 2 == tgtHi) v.z += 2.0f;
        if (col0 + 3 == tgtLo || col0 + 3 == tgtHi) v.w += 2.0f;
      }
    }
    o4[i] = v;
  }
}

extern "C" void kernel_launch(void* const* d_in, const int* in_sizes, int n_in,
                              void* d_out, int out_size, void* d_ws,
                              size_t ws_size, hipStream_t stream) {
  const float* x    = (const float*)d_in[0];
  const float* W1_0 = (const float*)d_in[1];
  const float* W2_0 = (const float*)d_in[2];
  const float* W3_0 = (const float*)d_in[3];
  const float* W1_1 = (const float*)d_in[4];
  const float* W2_1 = (const float*)d_in[5];
  const float* W3_1 = (const float*)d_in[6];
  int* ws = (int*)d_ws;  // T_TOK ints = 256 KB

  const size_t lds_bytes = (size_t)52608 * sizeof(float);
  bms_compute<<<N_TILES / 8, 256, lds_bytes, stream>>>(
      x, W1_0, W2_0, W3_0, W1_1, W2_1, W3_1, ws);
  bms_apply<<<2048, 256, 0, stream>>>(x, ws, (float*)d_out);
}